// ConvHull_Dictionnary_89386859364950
// MI455X (gfx1250) — compile-verified
//
#include <hip/hip_runtime.h>
#include <hip/hip_bf16.h>

typedef __attribute__((ext_vector_type(2))) float v2f;
typedef __attribute__((ext_vector_type(8))) float v8f;

#define NB_ATOMS   1024
#define FEAT       768
#define BATCH      16384
#define TEMP_F     100.0f
#define EPS_F      1e-8f

// ---------------------------------------------------------------------------
// fp32 WMMA GEMM: Out[M,N] = op(A)[M,K] * op(B)[K,N]  (+ compile-time epilogues)
//   TA: A stored transposed (A[k*lda + m]);  TB: B stored transposed (B[n*ldb + k])
//   SUBQ: Out -= Q;  DEN: atomicAdd(sum(Dm .* Out)) into *denom
//   CSCALE: scale B columns by colScale[n];  RSCALE: scale Out rows by rowScale[m]
// One wave -> 32x64 output tile: 2 A-fragments x 4 B-fragments -> 8 accumulators,
// 8 back-to-back v_wmma per K-step (B frags reused 2x, A frags 4x).
// ---------------------------------------------------------------------------
template<bool TA, bool TB, bool SUBQ, bool DEN, bool CSCALE, bool RSCALE>
__global__ __launch_bounds__(256)
void wmma_gemm(const float* __restrict__ A, int lda,
               const float* __restrict__ B, int ldb,
               const float* __restrict__ Q,
               const float* __restrict__ colScale,
               const float* __restrict__ rowScale,
               const float* __restrict__ Dm,
               float* __restrict__ Out, int ldo,
               float* __restrict__ denom,
               int M, int N, int K)
{
    const int lane  = threadIdx.x & 31;
    const int wave  = blockIdx.x * (blockDim.x >> 5) + (threadIdx.x >> 5);
    const int mt    = M >> 5;            // 32-row tiles
    const int nt    = N >> 6;            // 64-col tiles
    if (wave >= mt * nt) return;         // wave-uniform: EXEC stays all-ones
    const int m_tile = wave % mt;        // m-fastest: neighbors share B frags via L2
    const int n_tile = wave / mt;
    const int row0 = m_tile << 5;
    const int col0 = n_tile << 6;
    const int lr = lane & 15;            // row (A) / col (B,C) within 16
    const int lk = (lane >> 4) << 1;     // K base: 0 or 2

    v8f acc[2][4];
    #pragma unroll
    for (int s = 0; s < 2; ++s)
        #pragma unroll
        for (int t = 0; t < 4; ++t)
            acc[s][t] = {};

    float cs[4] = {1.f, 1.f, 1.f, 1.f};
    if (CSCALE) {
        #pragma unroll
        for (int t = 0; t < 4; ++t) cs[t] = colScale[col0 + 16 * t + lr];
    }

    #pragma unroll 2
    for (int k0 = 0; k0 < K; k0 += 4) {
        v2f a[2];
        if (TA) {
            const float* ap = A + (size_t)(k0 + lk) * lda + row0 + lr;
            a[0].x = ap[0];  a[0].y = ap[lda];
            a[1].x = ap[16]; a[1].y = ap[lda + 16];
        } else {
            const float* ap = A + (size_t)(row0 + lr) * lda + k0 + lk;
            a[0].x = ap[0];             a[0].y = ap[1];
            a[1].x = ap[16 * lda];      a[1].y = ap[16 * lda + 1];
        }
        v2f b[4];
        if (TB) {
            const float* bp = B + (size_t)(col0 + lr) * ldb + k0 + lk;
            #pragma unroll
            for (int t = 0; t < 4; ++t) {
                b[t].x = bp[16 * t * ldb];
                b[t].y = bp[16 * t * ldb + 1];
            }
        } else {
            const float* bp = B + (size_t)(k0 + lk) * ldb + col0 + lr;
            #pragma unroll
            for (int t = 0; t < 4; ++t) {
                b[t].x = bp[16 * t];
                b[t].y = bp[16 * t + ldb];
            }
        }
        if (CSCALE) {
            #pragma unroll
            for (int t = 0; t < 4; ++t) b[t] = b[t] * cs[t];
        }
        #pragma unroll
        for (int s = 0; s < 2; ++s)
            #pragma unroll
            for (int t = 0; t < 4; ++t)
                acc[s][t] = __builtin_amdgcn_wmma_f32_16x16x4_f32(
                    false, a[s], false, b[t], (short)0, acc[s][t], false, false);
    }

    // C/D layout: lane half selects M+0 / M+8 ; VGPR v -> row v (within 16-row sub-tile)
    const int mh = (lane >> 4) << 3;
    const int n0 = col0 + lr;
    float dpart = 0.f;
    #pragma unroll
    for (int s = 0; s < 2; ++s) {
        #pragma unroll
        for (int v = 0; v < 8; ++v) {
            const int m = row0 + 16 * s + v + mh;
            const size_t rowoff = (size_t)m * ldo;
            float rs = 1.f;
            if (RSCALE) rs = rowScale[m];
            #pragma unroll
            for (int t = 0; t < 4; ++t) {
                float o = acc[s][t][v];
                if (SUBQ) o -= Q[rowoff + n0 + 16 * t];
                if (RSCALE) o *= rs;
                Out[rowoff + n0 + 16 * t] = o;
                if (DEN) dpart += Dm[rowoff + n0 + 16 * t] * o;
            }
        }
    }
    if (DEN) {
        for (int off = 16; off; off >>= 1) dpart += __shfl_down(dpart, off, 32);
        if (lane == 0) atomicAdd(denom, dpart);
    }
}

// ---------------------------------------------------------------------------
// Prep kernels
// ---------------------------------------------------------------------------
__global__ __launch_bounds__(256) void ynorm_kernel(const float* __restrict__ y,
                                                    float* __restrict__ ynorm,
                                                    float* __restrict__ rnorm)
{
    const int n = blockIdx.x, tid = threadIdx.x;
    float s = 0.f;
    for (int k = tid; k < FEAT; k += 256) s += fabsf(y[(size_t)n * FEAT + k]);
    __shared__ float red[256];
    red[tid] = s; __syncthreads();
    for (int off = 128; off; off >>= 1) { if (tid < off) red[tid] += red[tid + off]; __syncthreads(); }
    if (tid == 0) { ynorm[n] = red[0]; rnorm[n] = 1.0f / red[0]; }
}

__global__ __launch_bounds__(256) void atom_norm_kernel(const float* __restrict__ atoms,
                                                        float* __restrict__ a_norm)
{
    const int i = blockIdx.x, tid = threadIdx.x;
    float s = 0.f;
    for (int k = tid; k < FEAT; k += 256) s += fabsf(atoms[(size_t)i * FEAT + k]);
    __shared__ float red[256];
    red[tid] = s; __syncthreads();
    for (int off = 128; off; off >>= 1) { if (tid < off) red[tid] += red[tid + off]; __syncthreads(); }
    __shared__ float rinv;
    if (tid == 0) rinv = 1.0f / red[0];
    __syncthreads();
    const float r = rinv;
    for (int k = tid; k < FEAT; k += 256) a_norm[(size_t)i * FEAT + k] = atoms[(size_t)i * FEAT + k] * r;
}

__global__ __launch_bounds__(256) void init_c_kernel(float* __restrict__ c)
{
    const float val = 1.0f / NB_ATOMS;
    for (int idx = blockIdx.x * 256 + threadIdx.x; idx < NB_ATOMS * BATCH; idx += gridDim.x * 256)
        c[idx] = val;
}

// ---------------------------------------------------------------------------
// Per-row argmin over the batch dim (nondiff step), first-min tie-break
// ---------------------------------------------------------------------------
__global__ __launch_bounds__(256) void argmin_row_kernel(const float* __restrict__ grad,
                                                         int* __restrict__ amcol)
{
    const int i = blockIdx.x, tid = threadIdx.x;
    float bv = INFINITY; int bj = BATCH;
    for (int j = tid; j < BATCH; j += 256) {
        const float v = grad[(size_t)i * BATCH + j];
        if (v < bv || (v == bv && j < bj)) { bv = v; bj = j; }
    }
    __shared__ float sv[256]; __shared__ int sj[256];
    sv[tid] = bv; sj[tid] = bj; __syncthreads();
    for (int off = 128; off; off >>= 1) {
        if (tid < off) {
            const float v2 = sv[tid + off]; const int j2 = sj[tid + off];
            if (v2 < sv[tid] || (v2 == sv[tid] && j2 < sj[tid])) { sv[tid] = v2; sj[tid] = j2; }
        }
        __syncthreads();
    }
    if (tid == 0) amcol[i] = sj[0];
}

// ---------------------------------------------------------------------------
// Nondiff: d = onehot(amcol) - c ; numer[j] = -sum_i d*grad (block owns 32 cols)
// ---------------------------------------------------------------------------
__global__ __launch_bounds__(256) void ndiff_d_kernel(const float* __restrict__ grad,
                                                      const float* __restrict__ c,
                                                      const int* __restrict__ amcol,
                                                      float* __restrict__ dbuf,
                                                      float* __restrict__ numer,
                                                      float* __restrict__ denom)
{
    const int tx = threadIdx.x & 31, ty = threadIdx.x >> 5;
    const int j = blockIdx.x * 32 + tx;
    if (blockIdx.x == 0 && threadIdx.x == 0) *denom = 0.f;   // consumed by next kernel
    float p = 0.f;
    for (int i = ty; i < NB_ATOMS; i += 8) {
        const size_t o = (size_t)i * BATCH + j;
        const float g = grad[o];
        const float dv = ((amcol[i] == j) ? 1.f : 0.f) - c[o];
        dbuf[o] = dv;
        p += dv * g;
    }
    __shared__ float red[256];
    red[threadIdx.x] = p; __syncthreads();
    for (int off = 4; off; off >>= 1) {
        if (ty < off) red[threadIdx.x] += red[threadIdx.x + off * 32];
        __syncthreads();
    }
    if (ty == 0) numer[j] = -red[tx];
}

// ---------------------------------------------------------------------------
// Diff pass 1: online column softmax stats of -TEMP*grad
// ---------------------------------------------------------------------------
__global__ __launch_bounds__(256) void softmax_ms_kernel(const float* __restrict__ grad,
                                                         float* __restrict__ mcol,
                                                         float* __restrict__ scol)
{
    const int tx = threadIdx.x & 31, ty = threadIdx.x >> 5;
    const int j = blockIdx.x * 32 + tx;
    float m = -INFINITY, s = 0.f;
    for (int i = ty; i < NB_ATOMS; i += 8) {
        const float x = -TEMP_F * grad[(size_t)i * BATCH + j];
        const float mn = fmaxf(m, x);
        s = s * __expf(m - mn) + __expf(x - mn);
        m = mn;
    }
    __shared__ float sm[256], ss[256];
    sm[threadIdx.x] = m; ss[threadIdx.x] = s; __syncthreads();
    for (int off = 4; off; off >>= 1) {
        if (ty < off) {
            const float m2 = sm[threadIdx.x + off * 32], s2 = ss[threadIdx.x + off * 32];
            const float mn = fmaxf(sm[threadIdx.x], m2);
            ss[threadIdx.x] = ss[threadIdx.x] * __expf(sm[threadIdx.x] - mn) + s2 * __expf(m2 - mn);
            sm[threadIdx.x] = mn;
        }
        __syncthreads();
    }
    if (ty == 0) { mcol[j] = sm[tx]; scol[j] = ss[tx]; }
}

// ---------------------------------------------------------------------------
// Diff pass 2: d = softmax - c ; numer[j] = -sum_i d*grad
// ---------------------------------------------------------------------------
__global__ __launch_bounds__(256) void diff_d_kernel(const float* __restrict__ grad,
                                                     const float* __restrict__ c,
                                                     const float* __restrict__ mcol,
                                                     const float* __restrict__ scol,
                                                     float* __restrict__ dbuf,
                                                     float* __restrict__ numer,
                                                     float* __restrict__ denom)
{
    const int tx = threadIdx.x & 31, ty = threadIdx.x >> 5;
    const int j = blockIdx.x * 32 + tx;
    if (blockIdx.x == 0 && threadIdx.x == 0) *denom = 0.f;
    const float mj = mcol[j], sinv = 1.0f / scol[j];
    float p = 0.f;
    for (int i = ty; i < NB_ATOMS; i += 8) {
        const size_t o = (size_t)i * BATCH + j;
        const float g = grad[o];
        const float sm_ = __expf(-TEMP_F * g - mj) * sinv;
        const float dv = sm_ - c[o];
        dbuf[o] = dv;
        p += dv * g;
    }
    __shared__ float red[256];
    red[threadIdx.x] = p; __syncthreads();
    for (int off = 4; off; off >>= 1) {
        if (ty < off) red[threadIdx.x] += red[threadIdx.x + off * 32];
        __syncthreads();
    }
    if (ty == 0) numer[j] = -red[tx];
}

// ---------------------------------------------------------------------------
// c += d * clamp(numer[j] / (denom + eps), 0, 1)
// ---------------------------------------------------------------------------
__global__ __launch_bounds__(256) void update_c_kernel(float* __restrict__ c,
                                                       const float* __restrict__ dbuf,
                                                       const float* __restrict__ numer,
                                                       const float* __restrict__ denom)
{
    const float den = *denom + EPS_F;
    for (int idx = blockIdx.x * 256 + threadIdx.x; idx < NB_ATOMS * BATCH; idx += gridDim.x * 256) {
        const int j = idx & (BATCH - 1);
        float lam = numer[j] / den;
        lam = fminf(fmaxf(lam, 0.f), 1.f);
        c[idx] += dbuf[idx] * lam;
    }
}

// ---------------------------------------------------------------------------
// Host orchestration
// ---------------------------------------------------------------------------
static inline int gemm_blocks(int M, int N) {
    const int waves = (M >> 5) * (N >> 6);
    return (waves * 32 + 255) / 256;
}

extern "C" void kernel_launch(void* const* d_in, const int* in_sizes, int n_in,
                              void* d_out, int out_size, void* d_ws, size_t ws_size,
                              hipStream_t stream) {
    (void)in_sizes; (void)n_in; (void)out_size; (void)ws_size;
    const float* y     = (const float*)d_in[0];   // [16384, 768]
    const float* atoms = (const float*)d_in[1];   // [1024, 768]
    float* out = (float*)d_out;                   // [16384, 768]

    char* w = (char*)d_ws;
    auto take = [&](size_t bytes) { void* p = (void*)w; w += (bytes + 255) & ~(size_t)255; return p; };
    float* a_norm = (float*)take(sizeof(float) * NB_ATOMS * FEAT);
    float* ynorm  = (float*)take(sizeof(float) * BATCH);
    float* rnorm  = (float*)take(sizeof(float) * BATCH);
    float* X      = (float*)take(sizeof(float) * NB_ATOMS * NB_ATOMS);
    float* q      = (float*)take(sizeof(float) * (size_t)NB_ATOMS * BATCH);
    float* c      = (float*)take(sizeof(float) * (size_t)NB_ATOMS * BATCH);
    float* grad   = (float*)take(sizeof(float) * (size_t)NB_ATOMS * BATCH);  // reused as Xd
    float* dbuf   = (float*)take(sizeof(float) * (size_t)NB_ATOMS * BATCH);
    float* numer  = (float*)take(sizeof(float) * BATCH);
    float* denom  = (float*)take(sizeof(float) * 64);
    float* mcol   = (float*)take(sizeof(float) * BATCH);
    float* scol   = (float*)take(sizeof(float) * BATCH);
    int*   amcol  = (int*)take(sizeof(int) * NB_ATOMS);

    // --- setup ---
    ynorm_kernel<<<BATCH, 256, 0, stream>>>(y, ynorm, rnorm);
    atom_norm_kernel<<<NB_ATOMS, 256, 0, stream>>>(atoms, a_norm);
    init_c_kernel<<<1024, 256, 0, stream>>>(c);

    // X = a_norm @ a_norm^T   (TB)
    wmma_gemm<false, true, false, false, false, false>
        <<<gemm_blocks(NB_ATOMS, NB_ATOMS), 256, 0, stream>>>(
        a_norm, FEAT, a_norm, FEAT, nullptr, nullptr, nullptr, nullptr,
        X, NB_ATOMS, nullptr, NB_ATOMS, NB_ATOMS, FEAT);

    // q = a_norm @ (y/|y|1)^T   (TB + column scale by 1/|y|1)
    wmma_gemm<false, true, false, false, true, false>
        <<<gemm_blocks(NB_ATOMS, BATCH), 256, 0, stream>>>(
        a_norm, FEAT, y, FEAT, nullptr, rnorm, nullptr, nullptr,
        q, BATCH, nullptr, NB_ATOMS, BATCH, FEAT);

    // --- 30 nondiff Frank-Wolfe steps ---
    for (int it = 0; it < 30; ++it) {
        wmma_gemm<false, false, true, false, false, false>
            <<<gemm_blocks(NB_ATOMS, BATCH), 256, 0, stream>>>(
            X, NB_ATOMS, c, BATCH, q, nullptr, nullptr, nullptr,
            grad, BATCH, nullptr, NB_ATOMS, BATCH, NB_ATOMS);
        argmin_row_kernel<<<NB_ATOMS, 256, 0, stream>>>(grad, amcol);
        ndiff_d_kernel<<<BATCH / 32, 256, 0, stream>>>(grad, c, amcol, dbuf, numer, denom);
        wmma_gemm<false, false, false, true, false, false>
            <<<gemm_blocks(NB_ATOMS, BATCH), 256, 0, stream>>>(
            X, NB_ATOMS, dbuf, BATCH, nullptr, nullptr, nullptr, dbuf,
            grad, BATCH, denom, NB_ATOMS, BATCH, NB_ATOMS);
        update_c_kernel<<<1024, 256, 0, stream>>>(c, dbuf, numer, denom);
    }

    // --- 10 diff (softmax) steps ---
    for (int it = 0; it < 10; ++it) {
        wmma_gemm<false, false, true, false, false, false>
            <<<gemm_blocks(NB_ATOMS, BATCH), 256, 0, stream>>>(
            X, NB_ATOMS, c, BATCH, q, nullptr, nullptr, nullptr,
            grad, BATCH, nullptr, NB_ATOMS, BATCH, NB_ATOMS);
        softmax_ms_kernel<<<BATCH / 32, 256, 0, stream>>>(grad, mcol, scol);
        diff_d_kernel<<<BATCH / 32, 256, 0, stream>>>(grad, c, mcol, scol, dbuf, numer, denom);
        wmma_gemm<false, false, false, true, false, false>
            <<<gemm_blocks(NB_ATOMS, BATCH), 256, 0, stream>>>(
            X, NB_ATOMS, dbuf, BATCH, nullptr, nullptr, nullptr, dbuf,
            grad, BATCH, denom, NB_ATOMS, BATCH, NB_ATOMS);
        update_c_kernel<<<1024, 256, 0, stream>>>(c, dbuf, numer, denom);
    }

    // recon = (c^T @ a_norm) * ynorm   (TA + row scale, straight to d_out)
    wmma_gemm<true, false, false, false, false, true>
        <<<gemm_blocks(BATCH, FEAT), 256, 0, stream>>>(
        c, BATCH, a_norm, FEAT, nullptr, nullptr, ynorm, nullptr,
        out, FEAT, nullptr, BATCH, FEAT, NB_ATOMS);
}